// DotProductDistributionHead_27556510171583
// MI455X (gfx1250) — compile-verified
//
#include <hip/hip_runtime.h>
#include <math.h>

typedef __attribute__((ext_vector_type(2))) float v2f;
typedef __attribute__((ext_vector_type(8))) float v8f;

#define DMODEL 128
#define NCAND  200
#define BATCH  4096
#define NPW    (NCAND / 8)   // candidates per wave = 25

__device__ __forceinline__ float gelu_erf(float v) {
    // exact (erf) GELU, matching jax.nn.gelu(approximate=False)
    return 0.5f * v * (1.0f + erff(v * 0.70710678118654752440f));
}

// -----------------------------------------------------------------------------
// Kernel 1: h = gelu(x @ W + b)  via V_WMMA_F32_16X16X4_F32 (exact f32 path)
//   grid.x = BATCH/16 blocks, 256 threads = 8 waves
//   wave w computes the 16x16 tile rows [16*blockIdx.x, +16), cols [16w, +16)
// -----------------------------------------------------------------------------
__global__ void __launch_bounds__(256)
mu_head_wmma_kernel(const float* __restrict__ x,
                    const float* __restrict__ W,     // [K=128][N=128] row-major
                    const float* __restrict__ bias,  // [128]
                    float* __restrict__ h)           // [BATCH][128]
{
    const int lane = threadIdx.x & 31;
    const int wave = threadIdx.x >> 5;
    const int m0   = blockIdx.x << 4;   // batch-row tile base
    const int n0   = wave << 4;         // output-col tile base (8 waves * 16 = 128)
    const int mr   = lane & 15;         // M (for A) / N (for B,C,D) within tile
    const int half = lane >> 4;         // 0: K={0,1}, 1: K={2,3}

    v8f acc = {0.f, 0.f, 0.f, 0.f, 0.f, 0.f, 0.f, 0.f};

    const float* __restrict__ xrow = x + (size_t)(m0 + mr) * DMODEL;

    #pragma unroll 8
    for (int k = 0; k < DMODEL; k += 4) {
        const int kb = k + (half << 1);
        // A 16x4 f32: lane m holds {A[m][kb], A[m][kb+1]} (kb even -> 8B aligned)
        v2f a = *(const v2f*)(xrow + kb);
        // B 4x16 f32: lane n holds {W[kb][n0+n], W[kb+1][n0+n]}
        v2f b;
        b.x = W[(size_t)kb * DMODEL + n0 + mr];
        b.y = W[(size_t)(kb + 1) * DMODEL + n0 + mr];
        acc = __builtin_amdgcn_wmma_f32_16x16x4_f32(
                  /*neg_a=*/false, a, /*neg_b=*/false, b,
                  /*c_mod=*/(short)0, acc, /*reuse_a=*/false, /*reuse_b=*/false);
    }

    // C/D layout: col = n0+mr; row = m0 + vgpr + 8*half
    const float bn = bias[n0 + mr];
    #pragma unroll
    for (int i = 0; i < 8; ++i) {
        const int row = m0 + i + (half << 3);
        h[(size_t)row * DMODEL + n0 + mr] = gelu_erf(acc[i] + bn);
    }
}

// -----------------------------------------------------------------------------
// Kernel 2: logits[b,n] = dot(h[b], emb[cand[b,n]]) + mu_bias[cand[b,n]]
//   grid.x = BATCH blocks, 256 threads = 8 waves.
//   Wave w owns the contiguous candidate chunk [w*25, (w+1)*25): sequential
//   index loads + sequential lane-0 stores. Constant 25-trip loop, unrolled x5
//   so ~5 fully-coalesced 512B embedding-row fetches are in flight per wave
//   (hides L2 latency; table is L2-resident at 51MB vs 192MB L2).
// -----------------------------------------------------------------------------
__global__ void __launch_bounds__(256)
gather_dot_kernel(const float* __restrict__ h,        // [BATCH][128]
                  const int*   __restrict__ cand,     // [BATCH][NCAND]
                  const float* __restrict__ mu_bias,  // [V]
                  const float* __restrict__ emb,      // [V][128]
                  float*       __restrict__ out)      // [BATCH][NCAND]
{
    const int b    = blockIdx.x;
    const int lane = threadIdx.x & 31;
    const int wave = threadIdx.x >> 5;

    // register-resident h[b]: lane holds floats [4*lane, 4*lane+4)
    const float4 hv = *(const float4*)(h + (size_t)b * DMODEL + (lane << 2));

    const int*  __restrict__ crow = cand + (size_t)b * NCAND + wave * NPW;
    float*      __restrict__ orow = out  + (size_t)b * NCAND + wave * NPW;

    #pragma unroll 5
    for (int t = 0; t < NPW; ++t) {
        // candidate index is wave-uniform -> put it in an SGPR (saddr path)
        const int idx = __builtin_amdgcn_readfirstlane(crow[t]);
        const float* __restrict__ erow = emb + (size_t)idx * DMODEL;

        const float4 e = *(const float4*)(erow + (lane << 2));

        float p = hv.x * e.x;
        p = fmaf(hv.y, e.y, p);
        p = fmaf(hv.z, e.z, p);
        p = fmaf(hv.w, e.w, p);

        // wave32 butterfly reduction
        p += __shfl_xor(p, 16, 32);
        p += __shfl_xor(p, 8, 32);
        p += __shfl_xor(p, 4, 32);
        p += __shfl_xor(p, 2, 32);
        p += __shfl_xor(p, 1, 32);

        if (lane == 0) {
            // NT store: don't let the 3.3MB of outputs evict L2-resident emb rows
            __builtin_nontemporal_store(p + mu_bias[idx], orow + t);
        }
    }
}

// -----------------------------------------------------------------------------
// launch
// inputs (setup_inputs order): 0:x 1:candidates 2:W_mu 3:b_mu 4:mu_bias 5:emb_table
// -----------------------------------------------------------------------------
extern "C" void kernel_launch(void* const* d_in, const int* in_sizes, int n_in,
                              void* d_out, int out_size, void* d_ws, size_t ws_size,
                              hipStream_t stream) {
    const float* x        = (const float*)d_in[0];
    const int*   cand     = (const int*)  d_in[1];
    const float* W_mu     = (const float*)d_in[2];
    const float* b_mu     = (const float*)d_in[3];
    const float* mu_bias  = (const float*)d_in[4];
    const float* emb      = (const float*)d_in[5];
    float*       out      = (float*)d_out;

    float* h = (float*)d_ws;   // BATCH * DMODEL floats = 2 MB scratch

    mu_head_wmma_kernel<<<BATCH / 16, 256, 0, stream>>>(x, W_mu, b_mu, h);
    gather_dot_kernel<<<BATCH, 256, 0, stream>>>(h, cand, mu_bias, emb, out);
}